// ExternalGraphConvolutionLayer_36112085025453
// MI455X (gfx1250) — compile-verified
//
#include <hip/hip_runtime.h>

typedef float v2f __attribute__((ext_vector_type(2)));
typedef float v8f __attribute__((ext_vector_type(8)));

#define DD 128

// ---------------------------------------------------------------------------
// Kernel 1: dual GEMM.  accU = H @ U  (written to d_out, becomes accumulator)
//                       hv   = H @ V  (written to workspace)
// One wave -> one 16x16 output tile, fp32 WMMA 16x16x4, K = 128 (32 steps).
// Block = 256 threads = 8 waves covering n = 0..127 of a 16-row strip.
// ---------------------------------------------------------------------------
__global__ void __launch_bounds__(256) gemm_dual(
    const float* __restrict__ H, const float* __restrict__ U,
    const float* __restrict__ V, float* __restrict__ accU,
    float* __restrict__ hv, int N) {
  const int lane  = threadIdx.x & 31;
  const int wave  = threadIdx.x >> 5;
  const int m0    = blockIdx.x * 16;
  const int n0    = wave * 16;
  const int ln    = lane & 15;           // M index for A, N index for B/D
  const int khalf = (lane >> 4) << 1;    // lanes 0-15 -> K+0/K+1, 16-31 -> K+2/K+3

  int arow = m0 + ln;
  if (arow >= N) arow = N - 1;           // clamp (no divergence; min-style)
  const float* Ap = H + (size_t)arow * DD + khalf;

  v8f cu = {};
  v8f cv = {};

#pragma unroll
  for (int k0 = 0; k0 < DD; k0 += 4) {
    // A fragment: A[m][k0+khalf], A[m][k0+khalf+1]  (8B-aligned pair)
    v2f a = *(const v2f*)(Ap + k0);

    // B fragments: B[k0+khalf][n], B[k0+khalf+1][n]
    const int kb = (k0 + khalf) * DD + n0 + ln;
    v2f bu, bv;
    bu[0] = U[kb];      bu[1] = U[kb + DD];
    bv[0] = V[kb];      bv[1] = V[kb + DD];

    cu = __builtin_amdgcn_wmma_f32_16x16x4_f32(false, a, false, bu,
                                               (short)0, cu, false, false);
    cv = __builtin_amdgcn_wmma_f32_16x16x4_f32(false, a, false, bv,
                                               (short)0, cv, false, false);
  }

  // D layout: VGPR r -> M = r (lanes 0-15) / r+8 (lanes 16-31), N = ln
  const int rbase = m0 + ((lane >> 4) << 3);
  const int col   = n0 + ln;
#pragma unroll
  for (int r = 0; r < 8; ++r) {
    const int rowo = rbase + r;
    if (rowo < N) {
      accU[(size_t)rowo * DD + col] = cu[r];
      hv  [(size_t)rowo * DD + col] = cv[r];
    }
  }
}

// ---------------------------------------------------------------------------
// Kernel 2: edge scatter.  acc[dst] += hv[src]  via native f32 global atomics.
// 32 lanes per edge, float4 per lane -> 4 hardware atomic adds per lane.
// 256 threads/block -> 8 edges per block.
// ---------------------------------------------------------------------------
__global__ void __launch_bounds__(256) scatter_edges(
    const float* __restrict__ hv, const int* __restrict__ esrc,
    const int* __restrict__ edst, float* __restrict__ acc, int E) {
  const int e = blockIdx.x * 8 + (threadIdx.x >> 5);
  if (e >= E) return;
  const int t4 = (threadIdx.x & 31) << 2;   // 4 floats per lane

  const int s = esrc[e];
  const int d = edst[e];

  const float4 vv = *(const float4*)(hv + (size_t)s * DD + t4);
  float* p = acc + (size_t)d * DD + t4;

  // Native L2 atomics (no return -> STOREcnt; S_ENDPGM waits idle).
  asm volatile(
      "global_atomic_add_f32 %0, %1, off\n\t"
      "global_atomic_add_f32 %0, %2, off offset:4\n\t"
      "global_atomic_add_f32 %0, %3, off offset:8\n\t"
      "global_atomic_add_f32 %0, %4, off offset:12"
      :
      : "v"(p), "v"(vv.x), "v"(vv.y), "v"(vv.z), "v"(vv.w)
      : "memory");
}

// ---------------------------------------------------------------------------
// Kernel 3: in-place ReLU on the accumulator (= d_out).
// ---------------------------------------------------------------------------
__global__ void __launch_bounds__(256) relu_inplace(float* __restrict__ x,
                                                    long long n) {
  long long i = (long long)blockIdx.x * blockDim.x + threadIdx.x;
  if (i < n) x[i] = fmaxf(x[i], 0.0f);
}

extern "C" void kernel_launch(void* const* d_in, const int* in_sizes, int n_in,
                              void* d_out, int out_size, void* d_ws, size_t ws_size,
                              hipStream_t stream) {
  const float* H    = (const float*)d_in[0];
  const float* U    = (const float*)d_in[1];
  const float* V    = (const float*)d_in[2];
  const int*   esrc = (const int*)d_in[3];
  const int*   edst = (const int*)d_in[4];

  const int N = in_sizes[0] / DD;     // 50000
  const int E = in_sizes[3];          // 600000

  float* out = (float*)d_out;         // doubles as the aggregation accumulator
  float* hv  = (float*)d_ws;          // N*DD floats = 25.6 MB scratch

  const int mtiles = (N + 15) / 16;   // 3125
  gemm_dual<<<mtiles, 256, 0, stream>>>(H, U, V, out, hv, N);

  scatter_edges<<<(E + 7) / 8, 256, 0, stream>>>(hv, esrc, edst, out, E);

  const long long total = (long long)N * DD;
  relu_inplace<<<(int)((total + 255) / 256), 256, 0, stream>>>(out, total);
}